// SimpleImprovedDilatedAttention_8615704396433
// MI455X (gfx1250) — compile-verified
//
#include <hip/hip_runtime.h>

// SimpleImprovedDilatedAttention for MI455X (gfx1250, wave32).
// Per (batch,pos,group): 16-head x 64-dim attention across heads.
// Memory-bound (~8 FLOP/byte) -> keep fp32, use V_WMMA_F32_16X16X4_F32.

typedef float v2f __attribute__((ext_vector_type(2)));
typedef float v8f __attribute__((ext_vector_type(8)));

#define BATCH 2
#define SEQN  8192
#define HEADS 48
#define DIMS  64
#define NGRP  3
#define TILES_PER_POS 3
#define TOTAL_TILES (BATCH * SEQN * TILES_PER_POS)   // 49152, tile = 1024 contiguous floats
#define TPB 4          // tiles per block
#define THREADS 128    // 4 wave32

#define QK_PITCH 68    // 16x64 tile rows padded to 68 floats (bank-conflict-free)
#define AT_PITCH 20    // 16x16 attn rows padded to 20 floats

__global__ __launch_bounds__(THREADS)
void dilated_attn_phase1(const float* __restrict__ q,
                         const float* __restrict__ k,
                         const float* __restrict__ v,
                         float* __restrict__ out,
                         float* __restrict__ denom)
{
    __shared__ float lq[TPB][16][QK_PITCH];   // 17408 B
    __shared__ float lk[TPB][16][QK_PITCH];   // 17408 B
    __shared__ float lv[TPB][16][QK_PITCH];   // 17408 B
    __shared__ float lat[TPB][16][AT_PITCH];  //  5120 B   total 57344 B static LDS

    const int tid = threadIdx.x;
    const long long tileBase = (long long)blockIdx.x * TPB;

    // ---- cooperative coalesced stage-in: 4 tiles x 1024 floats per array ----
    const float4* gq = (const float4*)q + tileBase * 256;
    const float4* gk = (const float4*)k + tileBase * 256;
    const float4* gv = (const float4*)v + tileBase * 256;
#pragma unroll
    for (int i = 0; i < 8; ++i) {
        int f4  = tid + THREADS * i;           // 0..1023 float4s
        int t   = f4 >> 8;
        int rem = f4 & 255;
        int row = rem >> 4;
        int c   = (rem & 15) << 2;
        float4 a = gq[f4]; *(float4*)&lq[t][row][c] = a;
        float4 b = gk[f4]; *(float4*)&lk[t][row][c] = b;
        float4 cc = gv[f4]; *(float4*)&lv[t][row][c] = cc;
    }
    __syncthreads();

    const int wave = tid >> 5;
    const int lane = tid & 31;
    const int hL   = lane & 15;        // N index / M row index per ISA layouts
    const int hsel = lane >> 4;        // which half of the wave
    const long long T = tileBase + wave;
    const int b   = (int)(T / (SEQN * TILES_PER_POS));
    const int rem = (int)(T % (SEQN * TILES_PER_POS));
    const int pos = rem / TILES_PER_POS;
    const int grp = rem % TILES_PER_POS;

    // ---- S = Q * K^T (16x16, contraction over d=64, 16 chained f32 WMMAs) ----
    v8f s = {0.f, 0.f, 0.f, 0.f, 0.f, 0.f, 0.f, 0.f};
#pragma unroll
    for (int kk = 0; kk < 64; kk += 4) {
        int c = kk + 2 * hsel;
        // A (16x4 f32): lane hL holds Q[hL][c], Q[hL][c+1]
        v2f a = { lq[wave][hL][c], lq[wave][hL][c + 1] };
        // B (4x16 f32): lane hL holds K^T[c][hL] = K[hL][c] (symmetric indexing)
        v2f bm = { lk[wave][hL][c], lk[wave][hL][c + 1] };
        s = __builtin_amdgcn_wmma_f32_16x16x4_f32(false, a, false, bm,
                                                  (short)0, s, false, false);
    }

    // ---- scale + softmax over N (k-head axis) : shfl_xor within 16-lane halves ----
    float at[8];
#pragma unroll
    for (int r = 0; r < 8; ++r) {
        float x = s[r] * 0.125f;                      // 1/sqrt(64)
        float m = x;
        m = fmaxf(m, __shfl_xor(m, 1));
        m = fmaxf(m, __shfl_xor(m, 2));
        m = fmaxf(m, __shfl_xor(m, 4));
        m = fmaxf(m, __shfl_xor(m, 8));
        float p = __expf(x - m);
        float su = p;
        su += __shfl_xor(su, 1);
        su += __shfl_xor(su, 2);
        su += __shfl_xor(su, 4);
        su += __shfl_xor(su, 8);
        at[r] = p / su;
    }
    // C/D layout -> LDS so we can re-read in A layout (row M = r + 8*hsel, col N = hL)
#pragma unroll
    for (int r = 0; r < 8; ++r)
        lat[wave][r + 8 * hsel][hL] = at[r];

    // ---- X = A * V (16x64 as four 16x16 N-blocks, K_c = 16 each) ----
    v8f xacc[4];
#pragma unroll
    for (int j = 0; j < 4; ++j) {
        v8f xa = {0.f, 0.f, 0.f, 0.f, 0.f, 0.f, 0.f, 0.f};
#pragma unroll
        for (int kk = 0; kk < 16; kk += 4) {
            int c = kk + 2 * hsel;
            v2f a  = { lat[wave][hL][c], lat[wave][hL][c + 1] };           // attn[M=hL][k=c..c+1]
            v2f bm = { lv[wave][c][16 * j + hL], lv[wave][c + 1][16 * j + hL] }; // V[k][d=16j+hL]
            xa = __builtin_amdgcn_wmma_f32_16x16x4_f32(false, a, false, bm,
                                                       (short)0, xa, false, false);
        }
        xacc[j] = xa;
    }

    // ---- partial denom: sum over all 16 heads per d channel, atomic into ws ----
    // segment for group g: 1024<<g ; blocks per batch: 8>>g ; slot = b*(8>>g) + pos>>(10+g)
    const int slot = (b << (3 - grp)) + (pos >> (10 + grp));
#pragma unroll
    for (int j = 0; j < 4; ++j) {
        float cs = xacc[j][0] + xacc[j][1] + xacc[j][2] + xacc[j][3] +
                   xacc[j][4] + xacc[j][5] + xacc[j][6] + xacc[j][7];   // 8 heads in this half
        cs += __shfl_xor(cs, 16);                                       // all 16 heads
        if (hsel == 0)
            atomicAdd(&denom[(grp * 16 + slot) * DIMS + 16 * j + hL], cs);
    }

    // ---- stage X through LDS (reuse lq region, per-wave private) and store coalesced ----
#pragma unroll
    for (int j = 0; j < 4; ++j)
#pragma unroll
        for (int r = 0; r < 8; ++r)
            lq[wave][r + 8 * hsel][16 * j + hL] = xacc[j][r];

    float4* gout = (float4*)out + T * 256;
#pragma unroll
    for (int i = 0; i < 8; ++i) {
        int f4  = lane + 32 * i;                 // 0..255 float4s of this tile
        int row = f4 >> 4;
        int c   = (f4 & 15) << 2;
        gout[f4] = *(float4*)&lq[wave][row][c];
    }
}

// ---- phase 2: out = x / (denom + 1e-8) / 3, elementwise (float4) ----
__global__ __launch_bounds__(256)
void dilated_attn_phase2(float* __restrict__ out, const float* __restrict__ denom)
{
    long long i = (long long)blockIdx.x * 256 + threadIdx.x;   // float4 index
    int hi  = (int)(i % 768);              // 48 heads * 16 float4s per position
    int h   = hi >> 4;
    int d4  = (hi & 15) << 2;
    long long pg = i / 768;                // global position 0..16383
    int b   = (int)(pg >> 13);
    int pos = (int)(pg & 8191);
    int grp = h >> 4;
    int slot = (b << (3 - grp)) + (pos >> (10 + grp));

    const float* dn = &denom[(grp * 16 + slot) * DIMS + d4];
    float4 x = ((float4*)out)[i];
    x.x = x.x / ((dn[0] + 1e-8f) * 3.0f);
    x.y = x.y / ((dn[1] + 1e-8f) * 3.0f);
    x.z = x.z / ((dn[2] + 1e-8f) * 3.0f);
    x.w = x.w / ((dn[3] + 1e-8f) * 3.0f);
    ((float4*)out)[i] = x;
}

__global__ void zero_denom(float* denom)
{
    int i = blockIdx.x * 256 + threadIdx.x;
    if (i < NGRP * 16 * DIMS) denom[i] = 0.f;
}

extern "C" void kernel_launch(void* const* d_in, const int* in_sizes, int n_in,
                              void* d_out, int out_size, void* d_ws, size_t ws_size,
                              hipStream_t stream)
{
    const float* q = (const float*)d_in[0];
    const float* k = (const float*)d_in[1];
    const float* v = (const float*)d_in[2];
    float* out   = (float*)d_out;
    float* denom = (float*)d_ws;   // 3*16*64 floats = 12 KB

    zero_denom<<<12, 256, 0, stream>>>(denom);
    dilated_attn_phase1<<<TOTAL_TILES / TPB, THREADS, 0, stream>>>(q, k, v, out, denom);
    long long elems4 = (long long)BATCH * SEQN * HEADS * DIMS / 4;   // 12,582,912
    dilated_attn_phase2<<<(int)(elems4 / 256), 256, 0, stream>>>(out, denom);
}